// AdMSoftmaxLoss_19421842112659
// MI455X (gfx1250) — compile-verified
//
#include <hip/hip_runtime.h>
#include <stdint.h>

// Problem constants (match reference setup_inputs)
#define B_      4
#define C_      2048
#define T_      8192
#define W_      256            // t-columns per block (= threads per block, 8 waves)
#define R_      32             // rows (c values) per TDM chunk
#define CSPLIT  4              // split of the C axis across blockIdx.z
#define RPB     (C_ / CSPLIT)  // 512 rows per block
#define NCHUNK  (RPB / R_)     // 16 chunks per block

// S = 30, M = 0.4 ; work in log2 domain: v = S*log2(e)*x
#define K2_     43.28085122666891f   // 30 * log2(e)
#define BIAS2_  17.312340490667565f  // 30*0.4 * log2(e)
#define LN2_    0.6931471805599453f

typedef uint32_t tdm_v4u __attribute__((ext_vector_type(4)));
typedef int      tdm_v8i __attribute__((ext_vector_type(8)));
typedef int      tdm_v4i __attribute__((ext_vector_type(4)));

// Issue one TDM load: 2D tile (R_ rows x W_ floats) of the (C_,T_) f32 plane -> LDS.
// D# group0: count=1 | lds_addr | global_addr[56:0] | type=2
// D# group1: data_size=4B, tensor_dim0=T_, tensor_dim1=C_, tile_dim0=W_,
//            tile_dim1=R_ (tile_dim2=0 => 2D), tensor_dim0_stride=T_.
static __device__ __forceinline__ void tdm_issue(uint64_t gaddr, uint32_t lds_addr) {
  tdm_v4u g0;
  g0.x = 1u;                                                    // count=1 (valid user D#)
  g0.y = lds_addr;                                              // lds_addr (bytes)
  g0.z = (uint32_t)gaddr;                                       // global_addr[31:0]
  g0.w = ((uint32_t)(gaddr >> 32) & 0x01FFFFFFu) | (2u << 30);  // addr[56:32] | type=2

  tdm_v8i g1 = {
    (int)0x00020000u,                                             // workgroup_mask=0, data_size=2 (4B)
    (int)(((uint32_t)T_ & 0xFFFFu) << 16),                        // tensor_dim0 lo16 @ [63:48]
    (int)(((uint32_t)T_ >> 16) | (((uint32_t)C_ & 0xFFFFu) << 16)), // td0 hi16 | td1 lo16
    (int)(((uint32_t)C_ >> 16) | ((uint32_t)W_ << 16)),           // td1 hi16 | tile_dim0
    (int)R_,                                                      // tile_dim1=R_, tile_dim2=0
    (int)T_,                                                      // tensor_dim0_stride lo32 (elements)
    0, 0                                                          // stride0 hi16, stride1 = 0
  };
  tdm_v4i gz4 = {0, 0, 0, 0};
  tdm_v8i gz8 = {0, 0, 0, 0, 0, 0, 0, 0};
  // amdgpu-toolchain (clang-23) 6-arg form: (g0, g1, g2, g3, extra, cpol)
  __builtin_amdgcn_tensor_load_to_lds(g0, g1, gz4, gz4, gz8, 0);
}

// Stage 1: streaming log2-sum-exp2 partials over a C-split, TDM double-buffered.
__global__ __launch_bounds__(W_) void admsl_lse_part(
    const float* __restrict__ outp,
    float* __restrict__ m_part, float* __restrict__ s_part) {
  __shared__ float buf[2][R_ * W_];

  const int tid = threadIdx.x;
  const int t0  = blockIdx.x * W_;
  const int b   = blockIdx.y;
  const int z   = blockIdx.z;

  const uint64_t base =
      (uint64_t)(uintptr_t)outp +
      (((uint64_t)b * C_ + (uint64_t)z * RPB) * (uint64_t)T_ + (uint64_t)t0) * 4ull;
  const uint32_t lds0 = (uint32_t)(uintptr_t)&buf[0][0];  // generic-ptr low 32 bits == LDS offset
  const uint32_t lds1 = (uint32_t)(uintptr_t)&buf[1][0];

  float m = -3.0e38f;  // running max in log2 domain
  float s = 0.0f;      // running sum of exp2(v - m)

  for (int ch = 0; ch < NCHUNK; ++ch) {
    if (tid < 32) {  // wave 0 drives the TDM pipeline (wave-uniform branch)
      if (ch == 0)
        tdm_issue(base, lds0);
      if (ch + 1 < NCHUNK) {
        tdm_issue(base + (uint64_t)(ch + 1) * R_ * T_ * 4ull,
                  ((ch + 1) & 1) ? lds1 : lds0);
        __builtin_amdgcn_s_wait_tensorcnt(1);   // chunk ch complete, ch+1 in flight
      } else {
        __builtin_amdgcn_s_wait_tensorcnt(0);   // last chunk complete
      }
    }
    __syncthreads();  // release all waves to consume buf[ch&1]

    const float* lb = (ch & 1) ? &buf[1][0] : &buf[0][0];

    // pass 1: raw max over this chunk's rows (max(v) = K2 * max(x), K2 > 0)
    float mraw = -3.0e38f;
#pragma unroll
    for (int r = 0; r < R_; ++r)
      mraw = fmaxf(mraw, lb[r * W_ + tid]);
    const float mn = fmaxf(m, K2_ * mraw);

    // pass 2: sum exp2(K2*x - mn), 4-way partial sums for ILP
    float a0 = 0.f, a1 = 0.f, a2 = 0.f, a3 = 0.f;
#pragma unroll
    for (int r = 0; r < R_; r += 4) {
      a0 += __builtin_amdgcn_exp2f(__builtin_fmaf(lb[(r + 0) * W_ + tid], K2_, -mn));
      a1 += __builtin_amdgcn_exp2f(__builtin_fmaf(lb[(r + 1) * W_ + tid], K2_, -mn));
      a2 += __builtin_amdgcn_exp2f(__builtin_fmaf(lb[(r + 2) * W_ + tid], K2_, -mn));
      a3 += __builtin_amdgcn_exp2f(__builtin_fmaf(lb[(r + 3) * W_ + tid], K2_, -mn));
    }
    s = s * __builtin_amdgcn_exp2f(m - mn) + ((a0 + a1) + (a2 + a3));
    m = mn;

    __syncthreads();  // all waves done with buf[ch&1] before it is re-filled
  }

  const int idx = ((z * B_ + b) * T_) + t0 + tid;
  m_part[idx] = m;
  s_part[idx] = s;
}

// Stage 2: merge C-splits, apply the additive-margin label correction, block-reduce.
__global__ __launch_bounds__(W_) void admsl_merge(
    const float* __restrict__ outp, const long long* __restrict__ target,
    const float* __restrict__ m_part, const float* __restrict__ s_part,
    float* __restrict__ block_sums) {
  const int tid = threadIdx.x;
  const int t   = blockIdx.x * W_ + tid;
  const int b   = blockIdx.y;

  float m = -3.0e38f, s = 0.0f;
#pragma unroll
  for (int z = 0; z < CSPLIT; ++z) {
    const float mp = m_part[((z * B_ + b) * T_) + t];
    const float sp = s_part[((z * B_ + b) * T_) + t];
    const float mn = fmaxf(m, mp);
    s = s * __builtin_amdgcn_exp2f(m - mn) + sp * __builtin_amdgcn_exp2f(mp - mn);
    m = mn;
  }

  const long long tg = target[(size_t)b * T_ + t];
  const bool valid = (tg != -1ll);
  const int lbl = valid ? (int)tg : 0;

  const float xl = outp[((uint64_t)b * C_ + (uint64_t)lbl) * (uint64_t)T_ + (uint64_t)t];
  const float vl = K2_ * xl;
  // replace label term exp2(vl-m) with margin term exp2(vl-bias-m)
  const float smod = s - __builtin_amdgcn_exp2f(vl - m)
                       + __builtin_amdgcn_exp2f(vl - BIAS2_ - m);
  const float L = LN2_ * ((vl - BIAS2_) - m - __builtin_amdgcn_logf(smod)); // logf = log2
  const float negL = valid ? -L : 0.0f;
  const float cnt  = valid ? 1.0f : 0.0f;

  __shared__ float r0[W_], r1[W_];
  r0[tid] = negL;
  r1[tid] = cnt;
  __syncthreads();
#pragma unroll
  for (int off = W_ / 2; off > 0; off >>= 1) {
    if (tid < off) {
      r0[tid] += r0[tid + off];
      r1[tid] += r1[tid + off];
    }
    __syncthreads();
  }
  if (tid == 0) {
    block_sums[2 * (b * gridDim.x + blockIdx.x) + 0] = r0[0];
    block_sums[2 * (b * gridDim.x + blockIdx.x) + 1] = r1[0];
  }
}

// Stage 3: deterministic scalar finish (128 pairs; trivial work).
__global__ void admsl_final(const float* __restrict__ block_sums,
                            float* __restrict__ outv, int nblk) {
  if (threadIdx.x == 0 && blockIdx.x == 0) {
    float acc = 0.0f;
    for (int b = 0; b < B_; ++b) {
      float sn = 0.0f, sc = 0.0f;
      for (int i = 0; i < nblk; ++i) {
        sn += block_sums[2 * (b * nblk + i) + 0];
        sc += block_sums[2 * (b * nblk + i) + 1];
      }
      acc += sn / sc;
    }
    *outv = acc * (1.0f / (float)B_);
  }
}

extern "C" void kernel_launch(void* const* d_in, const int* in_sizes, int n_in,
                              void* d_out, int out_size, void* d_ws, size_t ws_size,
                              hipStream_t stream) {
  const float*     outp   = (const float*)d_in[0];      // (B, C, T) f32
  const long long* target = (const long long*)d_in[1];  // (B, T) i64

  float* m_part     = (float*)d_ws;                             // CSPLIT*B*T floats
  float* s_part     = m_part + (size_t)CSPLIT * B_ * T_;        // CSPLIT*B*T floats
  float* block_sums = s_part + (size_t)CSPLIT * B_ * T_;        // 2 * B * (T/W) floats

  dim3 g1(T_ / W_, B_, CSPLIT);
  admsl_lse_part<<<g1, W_, 0, stream>>>(outp, m_part, s_part);

  dim3 g2(T_ / W_, B_);
  admsl_merge<<<g2, W_, 0, stream>>>(outp, target, m_part, s_part, block_sums);

  admsl_final<<<1, 32, 0, stream>>>(block_sums, (float*)d_out, T_ / W_);
}